// SparseAbacusLayer_25194278158502
// MI455X (gfx1250) — compile-verified
//
#include <hip/hip_runtime.h>

// SparseAbacusLayer for MI455X (gfx1250):
//   B=128, N_in=65536, N_out=65536, degree=2.
//   Strategy: per workgroup, TDM-DMA one 256KB activation row into LDS
//   (tensor_load_to_lds, TENSORcnt), then do all random interp gathers from
//   LDS (ds_load_b32), write coalesced outputs. Whole problem is L2-resident.

#define N_IN    65536
#define N_OUT   65536
#define BATCH   128
#define SPLIT   2          // N_out slices per batch row -> 256 blocks
#define THREADS 1024       // 32 wave32s; 1 block per WGP (LDS-limited)

typedef __attribute__((ext_vector_type(4))) unsigned int u32x4;
typedef __attribute__((ext_vector_type(4))) int          i32x4;
typedef __attribute__((ext_vector_type(8))) int          i32x8;

__global__ __launch_bounds__(THREADS, 1)
void sparse_abacus_kernel(const float* __restrict__ act,
                          const float* __restrict__ sp,
                          float* __restrict__ out)
{
    __shared__ float row[N_IN];   // 256 KB of the WGP's 320 KB LDS

    const int b    = blockIdx.x / SPLIT;
    const int part = blockIdx.x % SPLIT;
    const float* grow = act + (size_t)b * N_IN;

#if defined(__gfx1250__) && __has_builtin(__builtin_amdgcn_tensor_load_to_lds)
    // Wave 0 issues the TDM load of the whole row, waits for completion,
    // then the workgroup barrier releases the consumer waves.
    if (threadIdx.x < 32) {
        unsigned long long ga = (unsigned long long)(uintptr_t)grow;
        unsigned lds_off = (unsigned)(uintptr_t)(void*)row; // low 32b = LDS byte offset

        // ---- D# Group 0 (128b): count | lds_addr | global_addr | type=2 ----
        u32x4 g0;
        g0.x = 1u;                                                   // count=1, user D#
        g0.y = lds_off;                                              // lds_addr (bytes)
        g0.z = (unsigned)(ga & 0xFFFFFFFFull);                       // global_addr[31:0]
        g0.w = (unsigned)((ga >> 32) & 0x1FFFFFFull) | 0x80000000u;  // [56:32] | type=2

        // ---- D# Group 1 (256b): 2-D tile 4096x16 of 4-byte elements --------
        i32x8 g1;
        g1[0] = 0x00020000;          // data_size = 2 (4 bytes); no multicast/pad
        g1[1] = (int)(4096u << 16);  // tensor_dim0[15:0] = 4096
        g1[2] = (int)(16u   << 16);  // tensor_dim1[15:0] = 16
        g1[3] = (int)(4096u << 16);  // tile_dim0 = 4096
        g1[4] = 16;                  // tile_dim1 = 16, tile_dim2 = 0
        g1[5] = 4096;                // tensor_dim0_stride[31:0] = 4096
        g1[6] = (int)(4096u << 16);  // tensor_dim1_stride[15:0] (unused for 2-D)
        g1[7] = 0;

        i32x4 g2 = {0, 0, 0, 0};     // 2-D tile: groups 2/3 unused
        i32x4 g3 = {0, 0, 0, 0};
  #if __clang_major__ >= 23
        i32x8 g4 = {0, 0, 0, 0, 0, 0, 0, 0};
        __builtin_amdgcn_tensor_load_to_lds(g0, g1, g2, g3, g4, 0);
  #else
        __builtin_amdgcn_tensor_load_to_lds(g0, g1, g2, g3, 0);
  #endif
        __builtin_amdgcn_s_wait_tensorcnt(0);
    }
    __syncthreads();
#else
    // Fallback: cooperative vectorized fill of LDS.
    for (int i = threadIdx.x; i < N_IN / 4; i += THREADS)
        ((float4*)row)[i] = ((const float4*)grow)[i];
    __syncthreads();
#endif

    const int nPer  = N_OUT / SPLIT;        // 32768 outputs per block
    const int nBase = part * nPer;
    const float2* sp2 = (const float2*)sp;  // (N_out, degree=2) as float2
    float* orow = out + (size_t)b * N_OUT;

    const float DXF = 1.0f / 65535.0f;          // grid spacing
    const float INV = 1.0f / (DXF + 1e-8f);     // 1/(dx+eps)

    for (int n = nBase + (int)threadIdx.x; n < nBase + nPer; n += THREADS) {
        __builtin_prefetch(&sp2[n + THREADS], 0, 0);   // global_prefetch_b8
        float2 s2 = sp2[n];
        float o = 1.0f;
        #pragma unroll
        for (int d = 0; d < 2; ++d) {
            float s = (d == 0) ? s2.x : s2.y;
            s = fminf(fmaxf(s, 0.0f), 1.0f);
            // searchsorted-left on uniform grid: idx = ceil(s*(N-1)) - 1, clamped
            int idx = (int)ceilf(s * 65535.0f) - 1;
            idx = idx < 0 ? 0 : (idx > N_IN - 2 ? N_IN - 2 : idx);
            float a0 = row[idx];       // ds_load_b32 from LDS
            float a1 = row[idx + 1];
            a0 = fminf(fmaxf(a0, 0.0f), 1.0f);   // clip(activations)
            a1 = fminf(fmaxf(a1, 0.0f), 1.0f);
            float xoff  = s - (float)idx * DXF;
            float y     = fmaf((a1 - a0) * INV, xoff, a0);
            o *= (1.0f - y);           // fuzzy NAND: prod(1 - y)
        }
        orow[n] = o;
    }
}

extern "C" void kernel_launch(void* const* d_in, const int* in_sizes, int n_in,
                              void* d_out, int out_size, void* d_ws, size_t ws_size,
                              hipStream_t stream) {
    const float* act = (const float*)d_in[0];   // (128, 65536) f32
    const float* sp  = (const float*)d_in[1];   // (65536, 2, 1) f32
    float* out = (float*)d_out;                 // (128, 65536) f32
    (void)in_sizes; (void)n_in; (void)out_size; (void)d_ws; (void)ws_size;

    dim3 grid(BATCH * SPLIT);
    dim3 block(THREADS);
    sparse_abacus_kernel<<<grid, block, 0, stream>>>(act, sp, out);
}